// DeConvolution_16441134809110
// MI455X (gfx1250) — compile-verified
//
#include <hip/hip_runtime.h>
#include <hip/hip_bf16.h>

#define G 128
#define KOFF 27
#define C_IN 128
#define C_OUT 128
#define N_IN 50000
#define N_OUT 100000
#define GRID_CELLS (G * G * G)
#define W_ELEMS (KOFF * C_IN * C_OUT)

typedef float v2f __attribute__((ext_vector_type(2)));
typedef float v4f __attribute__((ext_vector_type(4)));
typedef float v8f __attribute__((ext_vector_type(8)));

// ---------------------------------------------------------------------------
// Kernel 1: init voxel grid to -1
// ---------------------------------------------------------------------------
__global__ __launch_bounds__(256) void grid_fill_kernel(int* __restrict__ grid) {
    int i = blockIdx.x * blockDim.x + threadIdx.x;
    if (i < GRID_CELLS) grid[i] = -1;
}

// ---------------------------------------------------------------------------
// Kernel 2: scatter input point indices (max semantics, matches reference)
// ---------------------------------------------------------------------------
__global__ __launch_bounds__(256) void grid_scatter_kernel(
        const int* __restrict__ inp_pos, int* __restrict__ grid) {
    int i = blockIdx.x * blockDim.x + threadIdx.x;
    if (i < N_IN) {
        int x = inp_pos[3 * i + 0];
        int y = inp_pos[3 * i + 1];
        int z = inp_pos[3 * i + 2];
        atomicMax(&grid[(x * G + y) * G + z], i);
    }
}

// ---------------------------------------------------------------------------
// Kernel 3: one-time W relayout into WMMA-native, mirror-folded form.
//   Wt[((k*32 + kk)*2 + hi)*256 + 2*n + j] = W[26-k][4*kk + 2*hi + j][n]
// ---------------------------------------------------------------------------
__global__ __launch_bounds__(256) void w_transpose_kernel(
        const float* __restrict__ W, float* __restrict__ Wt) {
    int i = blockIdx.x * blockDim.x + threadIdx.x;
    if (i >= W_ELEMS) return;
    int n  = i & 127;
    int kc = (i >> 7) & 127;
    int k  = i >> 14;
    int kk  = kc >> 2;
    int rem = kc & 3;
    int hi  = rem >> 1;
    int j   = rem & 1;
    float v = W[((26 - k) * C_IN + kc) * C_OUT + n];
    Wt[(((k * 32 + kk) * 2 + hi) * 256) + 2 * n + j] = v;
}

// ---------------------------------------------------------------------------
// Async copy helpers (CDNA5 gfx1250): global -> LDS via ASYNCcnt path.
// Moves 32 bytes (one feature-row slice per thread) as two B128 transfers.
// INST_OFFSET adds to both the global and LDS addresses (ISA 08_async §4.4).
// ---------------------------------------------------------------------------
__device__ __forceinline__ void async_copy_32B(unsigned lds_off,
                                               const float* gsrc) {
    unsigned long long ga = (unsigned long long)(uintptr_t)gsrc;
    asm volatile("global_load_async_to_lds_b128 %0, %1, off offset:0"
                 :: "v"(lds_off), "v"(ga) : "memory");
    asm volatile("global_load_async_to_lds_b128 %0, %1, off offset:16"
                 :: "v"(lds_off), "v"(ga) : "memory");
}

__device__ __forceinline__ void wait_async0() {
    asm volatile("s_wait_asynccnt 0x0" ::: "memory");
}

// ---------------------------------------------------------------------------
// Kernel 4: main deconv. Block = 16 output points x 128 out channels.
// 8 waves; wave w computes N-tile [16w,16w+16). Live offsets compacted; LDS
// A-tile double-buffered; gather uses async global->LDS overlapped with WMMA.
// ---------------------------------------------------------------------------
template <bool USE_WT>
__global__ __launch_bounds__(256) void deconv_wmma_kernel(
        const float* __restrict__ features,
        const float* __restrict__ W,          // [27][128][128] (original)
        const float* __restrict__ Wt,         // relayout (if USE_WT)
        const int*   __restrict__ out_pos,    // [N_OUT][3]
        const int*   __restrict__ grid,       // [G^3]
        float*       __restrict__ out) {      // [N_OUT][128]

    __shared__ int   s_pos[48];
    __shared__ int   s_idx[KOFF][16];
    __shared__ int   s_any[KOFF];
    __shared__ int   s_list[KOFF];
    __shared__ int   s_cnt;
    __shared__ __align__(16) float s_a[2][16][132];  // double-buffered A tile

    const int tid   = threadIdx.x;
    const int lane  = tid & 31;
    const int wave  = tid >> 5;
    const int lo    = lane & 15;
    const int hi    = lane >> 4;
    const int mbase = blockIdx.x * 16;

    if (tid < KOFF) s_any[tid] = 0;
    if (tid < 48)   s_pos[tid] = out_pos[3 * mbase + tid];
    __syncthreads();

    // ---- neighbor lookup: 27 offsets x 16 points ----
    for (int t = tid; t < KOFF * 16; t += 256) {
        int k = t >> 4;
        int r = t & 15;
        int dx = k / 9 - 1;
        int dy = (k / 3) % 3 - 1;
        int dz = k % 3 - 1;
        int x = s_pos[3 * r + 0] + dx;
        int y = s_pos[3 * r + 1] + dy;
        int z = s_pos[3 * r + 2] + dz;
        int idx = -1;
        if (((unsigned)x < (unsigned)G) & ((unsigned)y < (unsigned)G) &
            ((unsigned)z < (unsigned)G)) {
            idx = grid[(x * G + y) * G + z];
        }
        s_idx[k][r] = idx;
        if (idx >= 0) atomicOr(&s_any[k], 1);
    }
    __syncthreads();

    if (tid == 0) {                            // compact live offsets
        int c = 0;
        for (int k = 0; k < KOFF; ++k)
            if (s_any[k]) s_list[c++] = k;
        s_cnt = c;
    }
    __syncthreads();
    const int cnt = s_cnt;

    // per-thread gather role: row gr, 8 floats at column gc0
    const int gr  = tid >> 4;
    const int gc0 = (tid & 15) * 8;
    // LDS byte offsets of this thread's slice in each buffer (flat-address
    // low 32 bits == LDS offset, per aperture mapping)
    const unsigned ldsoff0 =
        (unsigned)(uintptr_t)&s_a[0][gr][gc0];
    const unsigned ldsoff1 =
        (unsigned)(uintptr_t)&s_a[1][gr][gc0];

    auto issue_gather = [&](int li, int buf) {
        int idx = s_idx[s_list[li]][gr];
        unsigned dst = buf ? ldsoff1 : ldsoff0;
        if (idx >= 0) {
            async_copy_32B(dst, features + (long)idx * C_IN + gc0);
        } else {
            v4f z = {};
            *(v4f*)&s_a[buf][gr][gc0 + 0] = z;
            *(v4f*)&s_a[buf][gr][gc0 + 4] = z;
        }
    };

    v8f acc = {};                              // 16x16 f32 C/D tile
    const int n = wave * 16 + lo;

    if (cnt > 0) issue_gather(0, 0);           // prologue

    for (int i = 0; i < cnt; ++i) {
        wait_async0();                         // tile i copies landed in LDS
        __syncthreads();                       // publish tile i block-wide

        // issue tile i+1 into the other buffer (its previous readers all
        // finished before the barrier above)
        if (i + 1 < cnt) issue_gather(i + 1, (i + 1) & 1);

        const int k   = s_list[i];
        const int buf = i & 1;
        const float* wb;
        if constexpr (USE_WT) {
            wb = Wt + (long)k * 32 * 2 * 256;
        } else {
            wb = W + (long)(26 - k) * C_IN * C_OUT;
        }

#pragma unroll
        for (int kk = 0; kk < 32; ++kk) {
            int kb = 4 * kk + 2 * hi;
            v2f a, b;
            a.x = s_a[buf][lo][kb + 0];        // A 16x4: row=lo, K pair by hi
            a.y = s_a[buf][lo][kb + 1];
            if constexpr (USE_WT) {
                b = *(const v2f*)(wb + ((kk * 2 + hi) * 256) + 2 * n);
            } else {
                b.x = wb[(kb + 0) * C_OUT + n];
                b.y = wb[(kb + 1) * C_OUT + n];
            }
            acc = __builtin_amdgcn_wmma_f32_16x16x4_f32(
                false, a, false, b, (short)0, acc, false, false);
        }
    }

    // ---- store D: VGPR r -> M = r + 8*hi, N = n ----
#pragma unroll
    for (int r = 0; r < 8; ++r) {
        int m = mbase + r + 8 * hi;
        out[(long)m * C_OUT + n] = acc[r];
    }
}

// ---------------------------------------------------------------------------
extern "C" void kernel_launch(void* const* d_in, const int* in_sizes, int n_in,
                              void* d_out, int out_size, void* d_ws, size_t ws_size,
                              hipStream_t stream) {
    const float* features = (const float*)d_in[0];
    const float* W        = (const float*)d_in[1];
    const int*   inp_pos  = (const int*)d_in[2];
    const int*   out_pos  = (const int*)d_in[3];
    float*       out      = (float*)d_out;

    const size_t grid_bytes = (size_t)GRID_CELLS * sizeof(int);
    const size_t wt_bytes   = (size_t)W_ELEMS * sizeof(float);

    int*   grid = (int*)d_ws;
    float* Wt   = (float*)((char*)d_ws + grid_bytes);
    const bool use_wt = ws_size >= grid_bytes + wt_bytes;

    grid_fill_kernel<<<(GRID_CELLS + 255) / 256, 256, 0, stream>>>(grid);
    grid_scatter_kernel<<<(N_IN + 255) / 256, 256, 0, stream>>>(inp_pos, grid);

    if (use_wt) {
        w_transpose_kernel<<<(W_ELEMS + 255) / 256, 256, 0, stream>>>(W, Wt);
        deconv_wmma_kernel<true><<<N_OUT / 16, 256, 0, stream>>>(
            features, W, Wt, out_pos, grid, out);
    } else {
        deconv_wmma_kernel<false><<<N_OUT / 16, 256, 0, stream>>>(
            features, W, Wt, out_pos, grid, out);
    }
}